// SpeechModel_16149077033115
// MI455X (gfx1250) — compile-verified
//
#include <hip/hip_runtime.h>

#define H4     4
#define T_LEN  4000
#define BATCH  256
#define NCLS   29
#define NROWS  (BATCH * T_LEN)   // 1,024,000 rows of hs / output

typedef __attribute__((ext_vector_type(2))) float v2f;
typedef __attribute__((ext_vector_type(8))) float v8f;

__device__ __forceinline__ float fsigmoid(float x) {
  // 1 / (1 + e^-x) : v_exp_f32 + v_rcp_f32, plenty accurate for LSTM gates
  return __builtin_amdgcn_rcpf(1.0f + __expf(-x));
}

// ---------------------------------------------------------------------------
// Kernel 1: LSTM scan. 4 lanes per batch element (one hidden unit each).
// All weights/state in registers; h shared across the 4-lane group with
// shfl_xor, W_hh coefficients pre-permuted by xor distance so no reorder.
// Writes hs[row][j] (row = b*T + t) to workspace for the WMMA projection.
// ---------------------------------------------------------------------------
__global__ void lstm_scan_kernel(const float* __restrict__ x,
                                 const float* __restrict__ W_ih,   // (16,1)
                                 const float* __restrict__ W_hh,   // (16,4)
                                 const float* __restrict__ b_ih,   // (16,)
                                 const float* __restrict__ b_hh,   // (16,)
                                 float* __restrict__ hs)           // (NROWS,4)
{
  const int tid = blockIdx.x * blockDim.x + threadIdx.x;
  const int b = tid >> 2;    // batch element
  const int j = tid & 3;     // hidden unit owned by this lane
  if (b >= BATCH) return;

  // Per-lane constants: gate slots s = {i,f,g,o} -> gate row g = s*4 + j
  float win[4], ball[4], whh[4][4];
#pragma unroll
  for (int s = 0; s < 4; ++s) {
    const int g = s * 4 + j;
    win[s]  = W_ih[g];
    ball[s] = b_ih[g] + b_hh[g];
#pragma unroll
    for (int m = 0; m < 4; ++m)
      whh[s][m] = W_hh[g * 4 + (j ^ m)];   // coefficient for shfl_xor(h, m)
  }

  float h = 0.0f, c = 0.0f;
  const float* xb  = x  + (size_t)b * T_LEN;
  float*       hsb = hs + (size_t)b * T_LEN * 4 + j;

  for (int t = 0; t < T_LEN; ++t) {
    const float xt = xb[t];
    // share h within the 4-lane group (3 shuffles)
    const float v1 = __shfl_xor(h, 1);
    const float v2 = __shfl_xor(h, 2);
    const float v3 = __shfl_xor(v1, 2);

    float pre[4];
#pragma unroll
    for (int s = 0; s < 4; ++s) {
      float p = fmaf(xt, win[s], ball[s]);
      p = fmaf(whh[s][0], h,  p);
      p = fmaf(whh[s][1], v1, p);
      p = fmaf(whh[s][2], v2, p);
      p = fmaf(whh[s][3], v3, p);
      pre[s] = p;
    }

    const float ig = fsigmoid(pre[0]);
    const float fg = fsigmoid(pre[1]);
    const float gg = fmaf(2.0f, fsigmoid(2.0f * pre[2]), -1.0f); // tanh
    const float og = fsigmoid(pre[3]);

    c = fmaf(fg, c, ig * gg);
    const float tc = fmaf(2.0f, fsigmoid(2.0f * c), -1.0f);      // tanh(c)
    h = og * tc;

    hsb[(size_t)t * 4] = h;
  }
}

// ---------------------------------------------------------------------------
// Kernel 2: out = hs @ W_fc.T + b_fc via V_WMMA_F32_16X16X4_F32 (K=4 exact).
// One wave -> 16 rows x 32 cols (two WMMAs, cols 16..31 zero-padded past 28).
// Accumulator initialized with b_fc so bias is free.
// ---------------------------------------------------------------------------
__global__ void fc_wmma_kernel(const float* __restrict__ hs,     // (NROWS,4)
                               const float* __restrict__ W_fc,   // (29,4)
                               const float* __restrict__ b_fc,   // (29,)
                               float* __restrict__ out)          // (NROWS,29)
{
  const int wave     = (blockIdx.x * blockDim.x + threadIdx.x) >> 5;
  const int lane     = threadIdx.x & 31;
  const int row_base = wave * 16;
  const int lm       = lane & 15;
  const int k0       = (lane < 16) ? 0 : 2;   // ISA A/B layout: half-wave K split

  // A tile: lane holds hs[row][k0], hs[row][k0+1]
  const int row = row_base + lm;
  v2f a;
  a.x = hs[(size_t)row * 4 + k0];
  a.y = hs[(size_t)row * 4 + k0 + 1];

#pragma unroll
  for (int tile = 0; tile < 2; ++tile) {
    const int n = tile * 16 + lm;            // output class column
    v2f bm;
    float bias;
    if (n < NCLS) {
      bm.x = W_fc[n * 4 + k0];               // B[k][n] = W_fc[n][k]
      bm.y = W_fc[n * 4 + k0 + 1];
      bias = b_fc[n];
    } else {
      bm.x = 0.0f; bm.y = 0.0f; bias = 0.0f;
    }

    v8f acc;
#pragma unroll
    for (int r = 0; r < 8; ++r) acc[r] = bias;

    // D = A(16x4) * B(4x16) + C  -> v_wmma_f32_16x16x4_f32
    acc = __builtin_amdgcn_wmma_f32_16x16x4_f32(
        /*neg_a=*/false, a, /*neg_b=*/false, bm,
        /*c_mod=*/(short)0, acc, /*reuse_a=*/false, /*reuse_b=*/false);

    if (n < NCLS) {
      const int m_off = (lane < 16) ? 0 : 8; // D layout: lanes 16-31 -> M=8..15
#pragma unroll
      for (int r = 0; r < 8; ++r) {
        const int m = row_base + m_off + r;
        out[(size_t)m * NCLS + n] = acc[r];
      }
    }
  }
}

// ---------------------------------------------------------------------------
extern "C" void kernel_launch(void* const* d_in, const int* in_sizes, int n_in,
                              void* d_out, int out_size, void* d_ws, size_t ws_size,
                              hipStream_t stream) {
  const float* x    = (const float*)d_in[0];
  const float* W_ih = (const float*)d_in[1];
  const float* W_hh = (const float*)d_in[2];
  const float* b_ih = (const float*)d_in[3];
  const float* b_hh = (const float*)d_in[4];
  const float* W_fc = (const float*)d_in[5];
  const float* b_fc = (const float*)d_in[6];
  float* out = (float*)d_out;

  float* hs = (float*)d_ws;   // NROWS * 4 floats = 16.4 MB

  // Scan: 256 batch * 4 lanes = 1024 threads
  lstm_scan_kernel<<<(BATCH * 4 + 255) / 256, 256, 0, stream>>>(
      x, W_ih, W_hh, b_ih, b_hh, hs);

  // Projection: 64,000 wave-tiles of 16 rows; 8 waves per 256-thread block
  const int n_tiles  = NROWS / 16;           // 64,000
  const int n_blocks = n_tiles / 8;          // 8,000
  fc_wmma_kernel<<<n_blocks, 256, 0, stream>>>(hs, W_fc, b_fc, out);
}